// PairRepresentation_86595130622111
// MI455X (gfx1250) — compile-verified
//
#include <hip/hip_runtime.h>

typedef float v2f __attribute__((ext_vector_type(2)));
typedef float v4f __attribute__((ext_vector_type(4)));
typedef float v8f __attribute__((ext_vector_type(8)));

#define D_MODEL 256
#define D_PAIR   64
#define L_SEQ   768
#define NREL_PAD 1536              // 2*L-1 = 1535, padded to tile multiple
#define E_ROW0  1281               // MAX_LEN - (L-1): first rel index actually used

// ---------------------------------------------------------------------------
// GEMM: out(768x64) = x(768x256) @ W(256x64) + b   — one wave per 16x16 tile,
// V_WMMA_F32_16X16X4_F32, K stepped by 4.
// ---------------------------------------------------------------------------
__global__ __launch_bounds__(32)
void k_proj_in(const float* __restrict__ x, const float* __restrict__ W,
               const float* __restrict__ b, float* __restrict__ out) {
    const int lane  = threadIdx.x;            // 0..31
    const int m0    = blockIdx.x * 16;        // 48 M-tiles
    const int n0    = blockIdx.y * 16;        // 4  N-tiles
    const int mrow  = m0 + (lane & 15);       // A: lanes 0-15 & 16-31 both M=0..15
    const int ncol  = n0 + (lane & 15);       // B: cols striped across lanes
    const int khalf = (lane >> 4) * 2;        // lanes>=16 hold K=2,3 of each step

    v8f acc = {};
    for (int k = 0; k < D_MODEL; k += 4) {
        v2f a, bb;
        a.x  = x[mrow * D_MODEL + k + khalf];
        a.y  = x[mrow * D_MODEL + k + khalf + 1];
        bb.x = W[(k + khalf    ) * D_PAIR + ncol];
        bb.y = W[(k + khalf + 1) * D_PAIR + ncol];
        acc = __builtin_amdgcn_wmma_f32_16x16x4_f32(
                  false, a, false, bb, (short)0, acc, false, false);
    }
    const float bias  = b[ncol];
    const int   mbase = m0 + (lane >> 4) * 8; // C/D: m = 8*(lane>>4) + r
#pragma unroll
    for (int r = 0; r < 8; ++r)
        out[(mbase + r) * D_PAIR + ncol] = acc[r] + bias;
}

// ---------------------------------------------------------------------------
// GEMM: C(Mx256) = A(Mx64) @ Wo(64x256) [+ bo]  — same wave-tile scheme, K=64.
// ---------------------------------------------------------------------------
__global__ __launch_bounds__(32)
void k_proj_out(const float* __restrict__ A, const float* __restrict__ Wo,
                const float* __restrict__ bo, float* __restrict__ C) {
    const int lane  = threadIdx.x;
    const int m0    = blockIdx.x * 16;
    const int n0    = blockIdx.y * 16;        // 16 N-tiles
    const int mrow  = m0 + (lane & 15);
    const int ncol  = n0 + (lane & 15);
    const int khalf = (lane >> 4) * 2;

    v8f acc = {};
#pragma unroll
    for (int k = 0; k < D_PAIR; k += 4) {
        v2f a, b;
        a.x = A[mrow * D_PAIR + k + khalf];
        a.y = A[mrow * D_PAIR + k + khalf + 1];
        b.x = Wo[(k + khalf    ) * D_MODEL + ncol];
        b.y = Wo[(k + khalf + 1) * D_MODEL + ncol];
        acc = __builtin_amdgcn_wmma_f32_16x16x4_f32(
                  false, a, false, b, (short)0, acc, false, false);
    }
    const float bias  = bo ? bo[ncol] : 0.0f;  // uniform branch, EXEC unchanged
    const int   mbase = m0 + (lane >> 4) * 8;
#pragma unroll
    for (int r = 0; r < 8; ++r)
        C[(size_t)(mbase + r) * D_MODEL + ncol] = acc[r] + bias;
}

// ---------------------------------------------------------------------------
// out[i,j,:] = Lp[i,:] + Rp[j,:] + Ep[j-i+767,:]   (bo already folded into Lp)
// Block: one i, 32 j's. 604 MB streamed out with non-temporal stores.
// ---------------------------------------------------------------------------
__global__ __launch_bounds__(256)
void k_bcast(const float* __restrict__ Lp, const float* __restrict__ Rp,
             const float* __restrict__ Ep, float* __restrict__ out) {
    const int i   = blockIdx.y;
    const int tid = threadIdx.x;
    const int k4  = tid & 63;                 // float4 slot within 256-wide row
    const int jr  = tid >> 6;                 // 0..3
    const v4f li  = ((const v4f*)(Lp + (size_t)i * D_MODEL))[k4];
    const int jbase = blockIdx.x * 32;

#pragma unroll
    for (int jj = 0; jj < 8; ++jj) {
        const int j = jbase + jj * 4 + jr;
        const int d = j - i + (L_SEQ - 1);    // 0..1534
        const v4f r = ((const v4f*)(Rp + (size_t)j * D_MODEL))[k4];
        const v4f e = ((const v4f*)(Ep + (size_t)d * D_MODEL))[k4];
        const v4f o = li + r + e;
        v4f* dst = (v4f*)(out + ((size_t)i * L_SEQ + j) * D_MODEL);
        __builtin_nontemporal_store(o, dst + k4);
    }
}

extern "C" void kernel_launch(void* const* d_in, const int* in_sizes, int n_in,
                              void* d_out, int out_size, void* d_ws, size_t ws_size,
                              hipStream_t stream) {
    const float* x  = (const float*)d_in[0];
    const float* Wl = (const float*)d_in[1];
    const float* bl = (const float*)d_in[2];
    const float* Wr = (const float*)d_in[3];
    const float* br = (const float*)d_in[4];
    const float* E  = (const float*)d_in[5];
    const float* Wo = (const float*)d_in[6];
    const float* bo = (const float*)d_in[7];
    float* out = (float*)d_out;

    float* ws    = (float*)d_ws;
    float* left  = ws;                          // 768*64
    float* right = left  + L_SEQ * D_PAIR;      // 768*64
    float* Lp    = right + L_SEQ * D_PAIR;      // 768*256
    float* Rp    = Lp    + L_SEQ * D_MODEL;     // 768*256
    float* Ep    = Rp    + L_SEQ * D_MODEL;     // 1536*256  (~3.4 MB total)

    // 1) left/right projections (WMMA f32)
    dim3 gA(L_SEQ / 16, D_PAIR / 16);
    k_proj_in<<<gA, 32, 0, stream>>>(x, Wl, bl, left);
    k_proj_in<<<gA, 32, 0, stream>>>(x, Wr, br, right);

    // 2) fold Wo (and bo into Lp) — tiny WMMA GEMMs
    dim3 gB(L_SEQ / 16, D_MODEL / 16);
    k_proj_out<<<gB, 32, 0, stream>>>(left,  Wo, bo,      Lp);
    k_proj_out<<<gB, 32, 0, stream>>>(right, Wo, nullptr, Rp);
    dim3 gE(NREL_PAD / 16, D_MODEL / 16);
    k_proj_out<<<gE, 32, 0, stream>>>(E + (size_t)E_ROW0 * D_PAIR, Wo, nullptr, Ep);

    // 3) bandwidth-bound broadcast add: 604 MB streamed out
    dim3 gC(L_SEQ / 32, L_SEQ);
    k_bcast<<<gC, 256, 0, stream>>>(Lp, Rp, Ep, out);
}